// CRFOutputLayer_18236431139090
// MI455X (gfx1250) — compile-verified
//
#include <hip/hip_runtime.h>
#include <hip/hip_bf16.h>

// ---------------------------------------------------------------------------
// CRF output layer for MI455X (gfx1250, wave32).
//   B=256, T=512, D=768, L=16
// Pipeline:
//   K1: logits = x @ W + b      -- WMMA f32 16x16x4, 1 wave per 16-row tile
//   K2: numerator per batch     -- 1 thread / batch
//   K3: forward logsumexp scan  -- 1 thread / (batch,state), shfl-exchange
//   K4: loss = -sum(num - logZ) -- single-block reduction
// ---------------------------------------------------------------------------

#define Bsz 256
#define Tsz 512
#define Dsz 768
#define Lsz 16

typedef float v2f __attribute__((ext_vector_type(2)));
typedef float v8f __attribute__((ext_vector_type(8)));

// ------------------------- K1: GEMM via f32 WMMA ---------------------------
// One wave computes a 16x16 tile: 16 consecutive (b,t) rows x all 16 labels.
// A (16x4 f32) layout: lanes 0-15 -> M=lane, VGPR0=K0,VGPR1=K1;
//                      lanes 16-31 -> M=lane-16, VGPR0=K2,VGPR1=K3.
// B (4x16 f32) symmetric: lane n holds column n, K pair selected by lane>=16.
__global__ __launch_bounds__(256) void gemm_logits_kernel(
    const float* __restrict__ x, const float* __restrict__ W,
    const float* __restrict__ bias, float* __restrict__ logits) {
  const int lane  = threadIdx.x & 31;
  const int wave  = threadIdx.x >> 5;
  const int m0    = (blockIdx.x * (blockDim.x >> 5) + wave) * 16;
  const int n     = lane & 15;      // column / A-row-within-tile
  const int khalf = lane >> 4;      // 0 -> K0,K1 ; 1 -> K2,K3

  const float* xr = x + (size_t)(m0 + n) * Dsz + 2 * khalf;
  const float* wc = W + n + 2 * khalf * Lsz;

  const float bn = bias[n];
  v8f acc = {bn, bn, bn, bn, bn, bn, bn, bn};

#pragma unroll 4
  for (int k = 0; k < Dsz; k += 4) {
    v2f a = *(const v2f*)(xr + k);                 // x[row][k+2*khalf .. +1]
    v2f bf;
    bf.x = wc[k * Lsz];                            // W[k+2*khalf    ][n]
    bf.y = wc[k * Lsz + Lsz];                      // W[k+2*khalf + 1][n]
    acc = __builtin_amdgcn_wmma_f32_16x16x4_f32(
        /*neg_a=*/false, a, /*neg_b=*/false, bf,
        /*c_mod=*/(short)0, acc, /*reuse_a=*/false, /*reuse_b=*/false);
  }

  // D layout: VGPR r, lanes 0-15 -> M=r ; lanes 16-31 -> M=8+r ; N = lane&15
  float* outp = logits + (size_t)m0 * Lsz;
  const int mbase = khalf * 8;
#pragma unroll
  for (int r = 0; r < 8; ++r)
    outp[(mbase + r) * Lsz + n] = acc[r];
}

// ------------------------- K2: numerator per batch -------------------------
__global__ __launch_bounds__(256) void crf_numerator_kernel(
    const float* __restrict__ logits, const int* __restrict__ mask,
    const int* __restrict__ labels, const float* __restrict__ transitions,
    const float* __restrict__ start_t, const float* __restrict__ end_t,
    float* __restrict__ numerator) {
  const int b = blockIdx.x * blockDim.x + threadIdx.x;
  if (b >= Bsz) return;
  const int*   lb = labels + b * Tsz;
  const int*   mk = mask + b * Tsz;
  const float* lg = logits + (size_t)b * Tsz * Lsz;

  float emit_score = 0.f, trans_score = 0.f;
  int count = 0;
  int prev = lb[0];
  for (int t = 0; t < Tsz; ++t) {
    const int m = mk[t];
    const int l = lb[t];
    const float mf = (m != 0) ? 1.0f : 0.0f;
    emit_score += mf * lg[t * Lsz + l];
    if (t > 0) trans_score += mf * transitions[prev * Lsz + l];
    prev = l;
    count += (m != 0) ? 1 : 0;
  }
  const int last_tag = lb[count - 1];
  numerator[b] = start_t[lb[0]] + emit_score + trans_score + end_t[last_tag];
}

// ------------------------- K3: forward scan --------------------------------
// tid = b*16 + j. Each wave holds 2 batches (lanes 0-15 and 16-31).
// alpha[j] lives in a register of lane j; cross-state reads via __shfl.
__global__ __launch_bounds__(256) void crf_scan_kernel(
    const float* __restrict__ logits, const int* __restrict__ mask,
    const float* __restrict__ transitions, const float* __restrict__ start_t,
    const float* __restrict__ end_t, float* __restrict__ log_z) {
  const int tid  = blockIdx.x * blockDim.x + threadIdx.x;  // 0..4095
  const int b    = tid >> 4;
  const int j    = tid & 15;
  const int lane = threadIdx.x & 31;
  const int half = lane & 16;  // which batch within the wave

  // Column j of the transition matrix: tr[i] = transitions[i][j]
  float tr[Lsz];
#pragma unroll
  for (int i = 0; i < Lsz; ++i) tr[i] = transitions[i * Lsz + j];

  const float* lg = logits + (size_t)b * Tsz * Lsz;
  const int*   mk = mask + b * Tsz;

  float alpha = start_t[j] + lg[j];

  for (int t = 1; t < Tsz; ++t) {
    const float emit = lg[t * Lsz + j];
    const int   m    = mk[t];

    float s[Lsz];
    float mx = -3.0e38f;
#pragma unroll
    for (int i = 0; i < Lsz; ++i) {
      const float ai = __shfl(alpha, half | i, 32);
      const float v  = ai + tr[i];
      s[i] = v;
      mx   = fmaxf(mx, v);
    }
    float acc = 0.f;
#pragma unroll
    for (int i = 0; i < Lsz; ++i) acc += __expf(s[i] - mx);
    const float na = emit + mx + __logf(acc);
    alpha = (m > 0) ? na : alpha;
  }

  // log_z[b] = logsumexp_j(alpha[j] + end_t[j]) via 16-lane xor reduction
  float v  = alpha + end_t[j];
  float mx = v;
#pragma unroll
  for (int off = 8; off >= 1; off >>= 1) mx = fmaxf(mx, __shfl_xor(mx, off, 32));
  float e = __expf(v - mx);
#pragma unroll
  for (int off = 8; off >= 1; off >>= 1) e += __shfl_xor(e, off, 32);
  if (j == 0) log_z[b] = mx + __logf(e);
}

// ------------------------- K4: final reduction -----------------------------
__global__ __launch_bounds__(256) void crf_reduce_kernel(
    const float* __restrict__ numerator, const float* __restrict__ log_z,
    float* __restrict__ out) {
  __shared__ float sm[Bsz];
  const int t = threadIdx.x;
  sm[t] = numerator[t] - log_z[t];
  __syncthreads();
  for (int s = Bsz / 2; s > 0; s >>= 1) {
    if (t < s) sm[t] += sm[t + s];
    __syncthreads();
  }
  if (t == 0) out[0] = -sm[0];
}

// ---------------------------------------------------------------------------
extern "C" void kernel_launch(void* const* d_in, const int* in_sizes, int n_in,
                              void* d_out, int out_size, void* d_ws, size_t ws_size,
                              hipStream_t stream) {
  const float* x       = (const float*)d_in[0];
  const int*   mask    = (const int*)d_in[1];
  const int*   labels  = (const int*)d_in[2];
  const float* W       = (const float*)d_in[3];
  const float* bias    = (const float*)d_in[4];
  const float* trans   = (const float*)d_in[5];
  const float* start_t = (const float*)d_in[6];
  const float* end_t   = (const float*)d_in[7];
  float*       out     = (float*)d_out;

  // Workspace layout
  char*  ws        = (char*)d_ws;
  float* logits    = (float*)ws;                                    // 8 MB
  float* numerator = (float*)(ws + (size_t)Bsz * Tsz * Lsz * 4);    // 1 KB
  float* log_z     = numerator + Bsz;                               // 1 KB

  // K1: 131072 rows / 16 per wave = 8192 waves; 8 waves per 256-thread block.
  gemm_logits_kernel<<<1024, 256, 0, stream>>>(x, W, bias, logits);

  // K2: one thread per batch.
  crf_numerator_kernel<<<1, 256, 0, stream>>>(logits, mask, labels, trans,
                                              start_t, end_t, numerator);

  // K3: 256 batches * 16 states = 4096 threads.
  crf_scan_kernel<<<16, 256, 0, stream>>>(logits, mask, trans, start_t, end_t,
                                          log_z);

  // K4: scalar loss.
  crf_reduce_kernel<<<1, 256, 0, stream>>>(numerator, log_z, out);
}